// EncodeStateDirectly_1803886265696
// MI455X (gfx1250) — compile-verified
//
#include <hip/hip_runtime.h>
#include <hip/hip_bf16.h>

// ---------------------------------------------------------------------------
// CDNA5 (gfx1250) implementation of the observability/least-squares loss.
// All heavy lifting is fp32 GEMM via V_WMMA_F32_16X16X4_F32 (wave32).
// ---------------------------------------------------------------------------

typedef __attribute__((ext_vector_type(2))) float v2f;
typedef __attribute__((ext_vector_type(8))) float v8f;

#define HIDN  256
#define OBSN  144
#define BATN  16
#define SEQN  256
#define SOBS  (SEQN * OBSN)     // 36864
#define MATE  (HIDN * HIDN)     // 65536
#define AUGC  272               // 256 + 16 rhs columns

__device__ __forceinline__ v8f wmma4(v2f a, v2f b, v8f c) {
  // 8 args: (neg_a, A, neg_b, B, c_mod, C, reuse_a, reuse_b)
  return __builtin_amdgcn_wmma_f32_16x16x4_f32(false, a, false, b, (short)0, c,
                                               false, false);
}

// ---------------------------------------------------------------------------
// Generic 64x64-tile fp32 WMMA GEMM.
//   C = op(A) * B  with optional per-COLUMN scale on B, batching, split-K
//   (atomicAdd outputs), and row/col guards (all dims multiple of 16 here).
// Block = 128 threads = 4 waves; each wave owns a 32x32 output patch
// (2x2 grid of 16x16 WMMA tiles, 64 K-steps of 4 per 64x64x256 tile).
// ---------------------------------------------------------------------------
__global__ __launch_bounds__(128)
void gemm_wmma(const float* __restrict__ A, const float* __restrict__ B,
               float* __restrict__ C,
               int M, int N, int K,
               long long sA, long long sB, long long sC,
               int lda, int ldb, int ldc,
               int transA, const float* __restrict__ colscale,
               int ksplit, int atomic_out)
{
  __shared__ float As[64][33];
  __shared__ float Bs[32][65];

  const int ks    = blockIdx.z % ksplit;
  const int batch = blockIdx.z / ksplit;
  const int klen  = (K + ksplit - 1) / ksplit;
  const int k0    = ks * klen;
  const int k1    = (k0 + klen < K) ? (k0 + klen) : K;

  const float* Ab = A + (long long)batch * sA;
  const float* Bb = B + (long long)batch * sB;
  float*       Cb = C + (long long)batch * sC;

  const int m0 = blockIdx.x * 64;
  const int n0 = blockIdx.y * 64;

  const int t    = threadIdx.x;
  const int wave = t >> 5;
  const int lane = t & 31;
  const int l    = lane & 15;
  const int h    = lane >> 4;
  const int wr   = (wave >> 1) * 32;
  const int wc   = (wave & 1) * 32;

  v8f acc[2][2];
#pragma unroll
  for (int i = 0; i < 2; ++i)
#pragma unroll
    for (int j = 0; j < 2; ++j)
      acc[i][j] = (v8f){0.f, 0.f, 0.f, 0.f, 0.f, 0.f, 0.f, 0.f};

  for (int kc = k0; kc < k1; kc += 32) {
    // hint the next chunk into cache
    if (kc + 32 < k1) {
      __builtin_prefetch(&Bb[(long long)(kc + 32) * ldb + n0], 0, 1);
      if (!transA) __builtin_prefetch(&Ab[(long long)m0 * lda + kc + 32], 0, 1);
    }
    // cooperative LDS staging: 2048 A elems + 2048 B elems, 16 each per thread
#pragma unroll
    for (int j = 0; j < 16; ++j) {
      const int idx = t + 128 * j;
      // A tile: [64 rows][32 k]
      {
        const int mm = idx >> 5, kk = idx & 31;
        const int gm = m0 + mm, gk = kc + kk;
        float va = 0.0f;
        if (gk < k1 && gm < M)
          va = transA ? Ab[(long long)gk * lda + gm]
                      : Ab[(long long)gm * lda + gk];
        As[mm][kk] = va;
      }
      // B tile: [32 k][64 cols]
      {
        const int kb = idx >> 6, nn = idx & 63;
        const int gk = kc + kb, gn = n0 + nn;
        float vb = 0.0f;
        if (gk < k1 && gn < N) {
          vb = Bb[(long long)gk * ldb + gn];
          if (colscale) vb *= colscale[gn];
        }
        Bs[kb][nn] = vb;
      }
    }
    __syncthreads();

#pragma unroll
    for (int kk = 0; kk < 32; kk += 4) {
      v2f af[2], bf[2];
#pragma unroll
      for (int mt = 0; mt < 2; ++mt) {
        af[mt].x = As[wr + 16 * mt + l][kk + 2 * h];
        af[mt].y = As[wr + 16 * mt + l][kk + 2 * h + 1];
      }
#pragma unroll
      for (int nt = 0; nt < 2; ++nt) {
        bf[nt].x = Bs[kk + 2 * h][wc + 16 * nt + l];
        bf[nt].y = Bs[kk + 2 * h + 1][wc + 16 * nt + l];
      }
#pragma unroll
      for (int mt = 0; mt < 2; ++mt)
#pragma unroll
        for (int nt = 0; nt < 2; ++nt)
          acc[mt][nt] = wmma4(af[mt], bf[nt], acc[mt][nt]);
    }
    __syncthreads();
  }

  // C/D layout: vgpr j holds M = j + 8*h, N = l  (16x16 f32 tile)
#pragma unroll
  for (int mt = 0; mt < 2; ++mt)
#pragma unroll
    for (int nt = 0; nt < 2; ++nt)
#pragma unroll
      for (int j = 0; j < 8; ++j) {
        const int gm = m0 + wr + 16 * mt + j + 8 * h;
        const int gn = n0 + wc + 16 * nt + l;
        if (gm < M && gn < N) {
          const float v = acc[mt][nt][j];
          const long long off = (long long)gm * ldc + gn;
          if (atomic_out) atomicAdd(&Cb[off], v);
          else Cb[off] = v;
        }
      }
}

// ---------------------------------------------------------------------------
// Fused first tree level: Out[i] = H(row i) @ H(row i+128)
//   H(u) = I - 2 uhat uhat^T;  product = I - 2aa' - 2bb' + 4(a.b) a b'
// One block per pair, thread t writes output row t.
// ---------------------------------------------------------------------------
__global__ __launch_bounds__(256)
void hh_pair(const float* __restrict__ R, float* __restrict__ Out)
{
  __shared__ float sa[256], sb[256];
  __shared__ float raa[256], rbb[256], rab[256];
  const int i = blockIdx.x, t = threadIdx.x;
  const float a = R[i * 256 + t];
  const float b = R[(i + 128) * 256 + t];
  sa[t] = a; sb[t] = b;
  raa[t] = a * a; rbb[t] = b * b; rab[t] = a * b;
  __syncthreads();
  for (int s = 128; s > 0; s >>= 1) {
    if (t < s) { raa[t] += raa[t + s]; rbb[t] += rbb[t + s]; rab[t] += rab[t + s]; }
    __syncthreads();
  }
  const float ia = rsqrtf(raa[0]);
  const float ib = rsqrtf(rbb[0]);
  const float d  = rab[0] * ia * ib;
  const float ar = a * ia;
  const float br = b * ib;
  float* orow = Out + (long long)i * MATE + (long long)t * 256;
  for (int c = 0; c < 256; ++c) {
    const float ac = sa[c] * ia, bc = sb[c] * ib;
    float v = -2.0f * ar * ac - 2.0f * br * bc + 4.0f * d * ar * bc;
    if (c == t) v += 1.0f;
    orow[c] = v;
  }
}

__global__ void svec_k(const float* __restrict__ alpha, float* __restrict__ S) {
  const int i = threadIdx.x;
  S[i] = -__cosf(alpha[i] * 0.5f) * 0.0f - cosf(alpha[i] * 0.5f); // = -cos(a/2)
}

__global__ void copy_k(const float* __restrict__ s, float* __restrict__ d, int n) {
  const int i = blockIdx.x * blockDim.x + threadIdx.x;
  if (i < n) d[i] = s[i];
}

__global__ void zero_k(float* __restrict__ d, int n) {
  const int i = blockIdx.x * blockDim.x + threadIdx.x;
  if (i < n) d[i] = 0.0f;
}

// Xmat[so*16 + b] = x[b*SOBS + so]  (= Y^T)
__global__ void xt_k(const float* __restrict__ x, float* __restrict__ Xm) {
  const int i = blockIdx.x * blockDim.x + threadIdx.x;
  if (i < SOBS * BATN) {
    const int so = i >> 4, b = i & 15;
    Xm[i] = x[(long long)b * SOBS + so];
  }
}

// Gauss-Jordan on augmented [G | rhs] (SPD -> no pivoting needed).
__global__ __launch_bounds__(256)
void solve_gj(float* __restrict__ Aug)
{
  __shared__ float piv[AUGC];
  const int t = threadIdx.x;
  for (int k = 0; k < 256; ++k) {
    const float inv = 1.0f / Aug[k * AUGC + k];
    for (int c = t; c < AUGC; c += 256) piv[c] = Aug[k * AUGC + c] * inv;
    __syncthreads();
    if (t == k) {
      for (int c = k; c < AUGC; ++c) Aug[k * AUGC + c] = piv[c];
    } else {
      const float f = Aug[t * AUGC + k];
      for (int c = k; c < AUGC; ++c) Aug[t * AUGC + c] -= f * piv[c];
    }
    __syncthreads();
  }
}

// sum over rows of || Y^T[row,:] - O[row,:] @ x0 ||^2  -> accum[0]
__global__ __launch_bounds__(256)
void resid_k(const float* __restrict__ O, const float* __restrict__ Xm,
             const float* __restrict__ Aug, float* __restrict__ accum)
{
  __shared__ float sx[256][16];
  __shared__ float red[256];
  const int t = threadIdx.x;
  for (int b = 0; b < 16; ++b) sx[t][b] = Aug[t * AUGC + 256 + b];
  __syncthreads();
  const int row = blockIdx.x * 256 + t;
  float acc[16];
#pragma unroll
  for (int b = 0; b < 16; ++b) acc[b] = 0.0f;
  const float* orow = O + (long long)row * 256;
  for (int k = 0; k < 256; ++k) {
    const float o = orow[k];
#pragma unroll
    for (int b = 0; b < 16; ++b) acc[b] += o * sx[k][b];
  }
  float loc = 0.0f;
#pragma unroll
  for (int b = 0; b < 16; ++b) {
    const float d = Xm[(long long)row * 16 + b] - acc[b];
    loc += d * d;
  }
  red[t] = loc;
  __syncthreads();
  for (int s = 128; s > 0; s >>= 1) {
    if (t < s) red[t] += red[t + s];
    __syncthreads();
  }
  if (t == 0) atomicAdd(accum, red[0]);
}

__global__ void logdet_k(const float* __restrict__ ld, float* __restrict__ accum) {
  __shared__ float red[256];
  const int t = threadIdx.x;
  float loc = 0.0f;
  for (int i = t; i < BATN * SEQN; i += 256) {
    const float v = fminf(ld[i], 0.0f);
    loc += v * v;
  }
  red[t] = loc;
  __syncthreads();
  for (int s = 128; s > 0; s >>= 1) {
    if (t < s) red[t] += red[t + s];
    __syncthreads();
  }
  if (t == 0) atomicAdd(accum + 1, red[0]);
}

__global__ void final_k(const float* __restrict__ accum, float* __restrict__ out) {
  out[0] = 0.5f * accum[0] / (float)(SOBS * BATN) +
           accum[1] / (float)(BATN * SEQN);
}

// ---------------------------------------------------------------------------

extern "C" void kernel_launch(void* const* d_in, const int* in_sizes, int n_in,
                              void* d_out, int out_size, void* d_ws, size_t ws_size,
                              hipStream_t stream)
{
  const float* x      = (const float*)d_in[0];   // [16,256,1,12,12]
  const float* U      = (const float*)d_in[1];   // [256,256]
  const float* V      = (const float*)d_in[2];   // [256,256]
  const float* alpha  = (const float*)d_in[3];   // [256]
  const float* Cmat   = (const float*)d_in[4];   // [144,256]
  const float* logdet = (const float*)d_in[5];   // [4096]
  float* out = (float*)d_out;

  float* w    = (float*)d_ws;
  float* bufA = w; w += (long long)128 * MATE;   // 32 MB
  float* bufB = w; w += (long long)64 * MATE;    // 16 MB
  float* Umat = w; w += MATE;
  float* Vtm  = w; w += MATE;
  float* Amat = w; w += MATE;
  float* Ap0  = w; w += MATE;
  float* Ap1  = w; w += MATE;
  float* Svec = w; w += 256;
  float* O    = w; w += (long long)SOBS * HIDN;  // 36 MB
  float* Xm   = w; w += (long long)SOBS * BATN;  // 2.25 MB
  float* Aug  = w; w += 256 * AUGC;              // accum follows directly
  float* accum = w; w += 8;

  auto gemm = [&](const float* A, const float* B, float* C, int M, int N, int K,
                  long long sA, long long sB, long long sC,
                  int lda, int ldb, int ldc, int transA, const float* cs,
                  int nbat, int ksplit, int atomic_out) {
    dim3 grid((M + 63) / 64, (N + 63) / 64, nbat * ksplit);
    gemm_wmma<<<grid, 128, 0, stream>>>(A, B, C, M, N, K, sA, sB, sC,
                                        lda, ldb, ldc, transA, cs,
                                        ksplit, atomic_out);
  };

  // ---- Householder product trees (pairing matches reference exactly) ----
  for (int which = 0; which < 2; ++which) {
    const float* rows = which ? V : U;
    float* dstmat = which ? Vtm : Umat;
    hh_pair<<<128, 256, 0, stream>>>(rows, bufA);     // 128 fused pair-products
    float* src = bufA;
    float* dst = bufB;
    for (int nb = 64; nb >= 1; nb >>= 1) {
      float* outp = (nb == 1) ? dstmat : dst;
      gemm(src, src + (long long)nb * MATE, outp, 256, 256, 256,
           MATE, MATE, MATE, 256, 256, 256, 0, nullptr, nb, 1, 0);
      float* tmp = src; src = dst; dst = tmp;
    }
  }

  // ---- A = Umat @ (Vt scaled per-COLUMN by S = -cos(alpha/2)) ----
  svec_k<<<1, 256, 0, stream>>>(alpha, Svec);
  gemm(Umat, Vtm, Amat, 256, 256, 256, 0, 0, 0, 256, 256, 256, 0, Svec, 1, 1, 0);

  // ---- O = [C; CA; ...; CA^255] via 8 doubling stages ----
  copy_k<<<(OBSN * HIDN + 255) / 256, 256, 0, stream>>>(Cmat, O, OBSN * HIDN);
  copy_k<<<(MATE + 255) / 256, 256, 0, stream>>>(Amat, Ap0, MATE);
  float* apc = Ap0;
  float* apn = Ap1;
  for (int k = 0; k < 8; ++k) {
    const int rows = OBSN << k;
    gemm(O, apc, O + (long long)rows * HIDN, rows, 256, 256,
         0, 0, 0, 256, 256, 256, 0, nullptr, 1, 1, 0);
    if (k < 7) {
      gemm(apc, apc, apn, 256, 256, 256, 0, 0, 0, 256, 256, 256,
           0, nullptr, 1, 1, 0);
      float* tmp = apc; apc = apn; apn = tmp;
    }
  }

  // ---- normal equations: Aug = [G | O^T Y^T], split-K atomic ----
  xt_k<<<(SOBS * BATN + 255) / 256, 256, 0, stream>>>(x, Xm);
  zero_k<<<(256 * AUGC + 8 + 255) / 256, 256, 0, stream>>>(Aug, 256 * AUGC + 8);
  gemm(O, O, Aug, 256, 256, SOBS, 0, 0, 0, 256, 256, AUGC,
       1, nullptr, 1, 64, 1);                       // G = O^T O
  gemm(O, Xm, Aug + 256, 256, 16, SOBS, 0, 0, 0, 256, 16, AUGC,
       1, nullptr, 1, 64, 1);                       // rhs = O^T Y^T

  // ---- solve + residual + loss ----
  solve_gj<<<1, 256, 0, stream>>>(Aug);
  resid_k<<<SOBS / 256, 256, 0, stream>>>(O, Xm, Aug, accum);
  logdet_k<<<1, 256, 0, stream>>>(logdet, accum);
  final_k<<<1, 1, 0, stream>>>(accum, out);
}